// variational_KFT_33586644254948
// MI455X (gfx1250) — compile-verified
//
#include <hip/hip_runtime.h>

// Problem constants (match reference)
#define B_TOTAL 131072
#define RNK     32
#define N0C     100000
#define N1C     50000
#define N2C     365

typedef float        v2f __attribute__((ext_vector_type(2)));
typedef float        v8f __attribute__((ext_vector_type(8)));
typedef unsigned int v4u __attribute__((ext_vector_type(4)));
typedef int          v4i __attribute__((ext_vector_type(4)));
typedef int          v8i __attribute__((ext_vector_type(8)));

#define WAVES_PER_BLOCK 4
#define THREADS (WAVES_PER_BLOCK * 32)
// TDM pad: 1 DWORD of LDS padding every 32 DWORDs -> row stride 33 (conflict-free)
#define TILE_STRIDE 33

// Issue one TDM 2D tile load: 32 rows x 32 fp32, row stride N1C*RNK elements,
// global start = base + ix*RNK*4, destination LDS offset lds_off, with LDS padding.
__device__ __forceinline__ void tdm_load_tile(unsigned lds_off, const float* gbase, int ix)
{
    unsigned long long ga = (unsigned long long)gbase + (unsigned long long)ix * (RNK * 4);

    v4u g0;
    g0.x = 1u;                                      // count=1, user descriptor
    g0.y = lds_off;                                 // LDS byte address
    g0.z = (unsigned)(ga & 0xFFFFFFFFu);            // global_addr[31:0]
    g0.w = (unsigned)((ga >> 32) & 0x1FFFFFFu)      // global_addr[56:32]
         | (2u << 30);                              // type=2 ("image")

    v8i g1;
    g1[0] = (2 << 16)      // data_size = 4 bytes
          | (1 << 20)      // pad_enable
          | (4 << 22);     // pad_interval: every 32 DWORDs (pad_amount=0 -> 1 DWORD)
    g1[1] = (RNK << 16);   // [47:32] barrier addr = 0 ; [63:48] tensor_dim0 lo16 = 32
    g1[2] = (RNK << 16);   // tensor_dim0 hi = 0 ; tensor_dim1 lo16 = 32
    g1[3] = (RNK << 16);   // tensor_dim1 hi = 0 ; tile_dim0 = 32
    g1[4] = RNK;           // tile_dim1 = 32 ; tile_dim2 = 0 (2D tile)
    g1[5] = N1C * RNK;     // tensor_dim0_stride = 1,600,000 elements (low 32)
    g1[6] = 0;             // stride hi16 = 0 ; tensor_dim1_stride = 0
    g1[7] = 0;

    v4i gz = {0, 0, 0, 0};
#if __clang_major__ >= 23
    v8i gz8 = {0, 0, 0, 0, 0, 0, 0, 0};
    __builtin_amdgcn_tensor_load_to_lds(g0, g1, gz, gz, gz8, 0);
#else
    __builtin_amdgcn_tensor_load_to_lds(g0, g1, gz, gz, 0);
#endif
}

__global__ __launch_bounds__(THREADS, 1)
void kft_bilinear_kernel(const int* __restrict__ idx,
                         const float* __restrict__ core0, const float* __restrict__ prime0,
                         const float* __restrict__ core1, const float* __restrict__ prime1,
                         const float* __restrict__ core2, const float* __restrict__ prime2,
                         float* __restrict__ out)
{
    __shared__ float t1[WAVES_PER_BLOCK][RNK * TILE_STRIDE];  // core1 tile (TDM dest)
    __shared__ float tp[WAVES_PER_BLOCK][RNK * TILE_STRIDE];  // prime1 tile (TDM dest)
    __shared__ float p0buf[WAVES_PER_BLOCK][RNK];
    __shared__ float p2buf[WAVES_PER_BLOCK][RNK];

    const int lane = threadIdx.x & 31;
    const int wv   = threadIdx.x >> 5;
    const int b    = blockIdx.x * WAVES_PER_BLOCK + wv;
    if (b >= B_TOTAL) return;

    // Wave-uniform indices (SGPRs) so the TDM descriptors stay scalar.
    const int ix0 = __builtin_amdgcn_readfirstlane(idx[b * 3 + 0]);
    const int ix1 = __builtin_amdgcn_readfirstlane(idx[b * 3 + 1]);
    const int ix2 = __builtin_amdgcn_readfirstlane(idx[b * 3 + 2]);

    // Kick off both 4.2KB tile DMAs (async, TENSORcnt) before doing anything else.
    tdm_load_tile((unsigned)(unsigned long long)&t1[wv][0], core1, ix1);
    tdm_load_tile((unsigned)(unsigned long long)&tp[wv][0], prime1, ix1);

    // Overlap: small gathers for p0 (contiguous 128B) and p2 (stride-365 floats).
    const float p0 = core0[ix0 * RNK + lane] * prime0[ix0 * RNK + lane];
    const float p2 = core2[lane * N2C + ix2] * prime2[lane * N2C + ix2];
    p0buf[wv][lane] = p0;
    p2buf[wv][lane] = p2;

    __builtin_amdgcn_s_wait_tensorcnt(0);
    __asm__ volatile("" ::: "memory");

    // y = (t1 .* tp) @ p2 via fp32 WMMA 16x16x4.
    // A frag (16x4 f32): lanes 0-15 row r K={c,c+1}; lanes 16-31 row r K={c+2,c+3}.
    // B frag: p2 K-chunk replicated across all 16 N columns -> every lane's C holds y.
    const int half = lane >> 4;          // 0: K offset 0, rows m%16<->VGPR m ; 1: K offset 2, m+8
    const int r    = lane & 15;
    const float* T1 = &t1[wv][0];
    const float* TP = &tp[wv][0];
    const float* P2 = &p2buf[wv][0];

    v8f c0 = {0, 0, 0, 0, 0, 0, 0, 0};  // y[0..15]
    v8f c1 = {0, 0, 0, 0, 0, 0, 0, 0};  // y[16..31]
    const int row0 = r * TILE_STRIDE;
    const int row1 = (16 + r) * TILE_STRIDE;
#pragma unroll
    for (int c = 0; c < RNK; c += 4) {
        const int k0 = c + 2 * half;
        v2f bf; bf.x = P2[k0];  bf.y = P2[k0 + 1];           // broadcast across N
        v2f a0; a0.x = T1[row0 + k0]     * TP[row0 + k0];
                a0.y = T1[row0 + k0 + 1] * TP[row0 + k0 + 1];
        v2f a1; a1.x = T1[row1 + k0]     * TP[row1 + k0];
                a1.y = T1[row1 + k0 + 1] * TP[row1 + k0 + 1];
        c0 = __builtin_amdgcn_wmma_f32_16x16x4_f32(false, a0, false, bf, (short)0, c0, false, false);
        c1 = __builtin_amdgcn_wmma_f32_16x16x4_f32(false, a1, false, bf, (short)0, c1, false, false);
    }

    // C layout: VGPR v, lane-half h -> row m = v + 8*h (all N columns identical).
    // out = sum_m p0[m] * y[m]; halves hold disjoint m-sets -> one shfl_xor(16).
    const float* P0 = &p0buf[wv][0];
    float part = 0.f;
#pragma unroll
    for (int v = 0; v < 8; ++v) {
        part += c0[v] * P0[v + 8 * half];
        part += c1[v] * P0[16 + v + 8 * half];
    }
    part += __shfl_xor(part, 16, 32);
    if (lane == 0) out[b] = part;
}

extern "C" void kernel_launch(void* const* d_in, const int* in_sizes, int n_in,
                              void* d_out, int out_size, void* d_ws, size_t ws_size,
                              hipStream_t stream)
{
    // setup_inputs() dict order: indices, core0, prime0, core1, prime1, core2, prime2
    const int*   idx    = (const int*)  d_in[0];
    const float* core0  = (const float*)d_in[1];
    const float* prime0 = (const float*)d_in[2];
    const float* core1  = (const float*)d_in[3];
    const float* prime1 = (const float*)d_in[4];
    const float* core2  = (const float*)d_in[5];
    const float* prime2 = (const float*)d_in[6];
    float* out = (float*)d_out;

    dim3 grid((B_TOTAL + WAVES_PER_BLOCK - 1) / WAVES_PER_BLOCK);
    kft_bilinear_kernel<<<grid, THREADS, 0, stream>>>(idx, core0, prime0, core1, prime1,
                                                      core2, prime2, out);
}